// interGConv_44530220925489
// MI455X (gfx1250) — compile-verified
//
#include <hip/hip_runtime.h>
#include <math.h>

typedef __attribute__((ext_vector_type(2))) float v2f;
typedef __attribute__((ext_vector_type(8))) float v8f;

#define Gc     5
#define Bc     8
#define CIN    320
#define Jc     256
#define Kc     62
#define CING   64
#define COUTG  96
#define OUTC   480
#define SP     15872                   /* Jc*Kc */
#define HSZ    ((size_t)Bc*OUTC*SP)    /* floats per intermediate tensor */
#define EPSBN  1e-5f

__device__ __forceinline__ v8f wmma_f32(v2f a, v2f b, v8f c) {
  // D = A(16x4) * B(4x16) + C, exact f32 matrix op on CDNA5
  return __builtin_amdgcn_wmma_f32_16x16x4_f32(false, a, false, b, (short)0, c, false, false);
}

// ---------------------------------------------------------------- utilities
__global__ void zero_stats_kernel(float* p, int n) {
  int i = blockIdx.x * blockDim.x + threadIdx.x;
  if (i < n) p[i] = 0.0f;
}

__global__ void bn_finalize_kernel(const float* __restrict__ sum,
                                   const float* __restrict__ sq,
                                   const float* __restrict__ gamma,
                                   const float* __restrict__ beta,
                                   float* __restrict__ scale,
                                   float* __restrict__ shift, float invN) {
  int c = blockIdx.x * blockDim.x + threadIdx.x;
  if (c < OUTC) {
    float m   = sum[c] * invN;
    float var = sq[c] * invN - m * m;
    float inv = rsqrtf(var + EPSBN);
    float s   = gamma[c] * inv;
    scale[c]  = s;
    shift[c]  = beta[c] - m * s;
  }
}

// ------------------------------------------------- conv1: x(64) -> h(96) + stats
// wave tile: 16 cout x 64 spatial; grid: G * 6 mtiles * B * 248 ntiles / 8 waves
__global__ __launch_bounds__(256)
void conv1_kernel(const float* __restrict__ x, const float* __restrict__ W1,
                  float* __restrict__ h, float* __restrict__ sum1,
                  float* __restrict__ sq1) {
  const int lane = threadIdx.x & 31;
  const int hl = lane >> 4, ln = lane & 15;
  int wid = blockIdx.x * 8 + (threadIdx.x >> 5);
  int g  = wid / 11904; int r = wid % 11904;   // 6*8*248
  int mt = r / 1984;    r %= 1984;             // 8*248
  int b  = r / 248;     int nt = r % 248;
  const int m0 = mt * 16;
  const int s0 = nt * 64;

  const float* Wg = W1 + (size_t)((g * COUTG) + m0 + ln) * CING;
  const float* xb = x + (size_t)(b * CIN + g * CING) * SP + s0 + ln;

  v8f acc[4] = {};
#pragma unroll
  for (int kk = 0; kk < 16; ++kk) {            // cin in steps of 4
    const int c = kk * 4 + hl * 2;
    v2f a; a.x = Wg[c]; a.y = Wg[c + 1];
    const float* bp = xb + (size_t)c * SP;
#pragma unroll
    for (int t = 0; t < 4; ++t) {
      v2f bb; bb.x = bp[t * 16]; bb.y = bp[t * 16 + SP];
      acc[t] = wmma_f32(a, bb, acc[t]);
    }
  }

  float* hb = h + (size_t)(b * OUTC + g * COUTG + m0 + hl * 8) * SP + s0 + ln;
#pragma unroll
  for (int v = 0; v < 8; ++v) {
    float s = 0.0f, q = 0.0f;
#pragma unroll
    for (int t = 0; t < 4; ++t) {
      float val = acc[t][v];
      hb[(size_t)v * SP + t * 16] = val;
      s += val; q += val * val;
    }
#pragma unroll
    for (int off = 8; off; off >>= 1) {        // reduce 16-lane half
      s += __shfl_xor(s, off, 16);
      q += __shfl_xor(q, off, 16);
    }
    if (ln == 0) {
      int ch = g * COUTG + m0 + hl * 8 + v;
      atomicAdd(&sum1[ch], s);
      atomicAdd(&sq1[ch], q);
    }
  }
}

// ------------------------------ conv2: BN1+ELU(h)(96) -> h2(96) + stats2
__global__ __launch_bounds__(256)
void conv2_kernel(const float* __restrict__ h, const float* __restrict__ W2,
                  const float* __restrict__ sc1, const float* __restrict__ sh1,
                  float* __restrict__ h2, float* __restrict__ sum2,
                  float* __restrict__ sq2) {
  const int lane = threadIdx.x & 31;
  const int hl = lane >> 4, ln = lane & 15;
  int wid = blockIdx.x * 8 + (threadIdx.x >> 5);
  int g  = wid / 11904; int r = wid % 11904;
  int mt = r / 1984;    r %= 1984;
  int b  = r / 248;     int nt = r % 248;
  const int m0 = mt * 16;
  const int s0 = nt * 64;

  const float* Wg = W2 + (size_t)((g * COUTG) + m0 + ln) * COUTG;
  const float* hbI = h + (size_t)(b * OUTC + g * COUTG) * SP + s0 + ln;

  v8f acc[4] = {};
#pragma unroll
  for (int kk = 0; kk < 24; ++kk) {            // cin(=96) in steps of 4
    const int c = kk * 4 + hl * 2;
    const int gch = g * COUTG + c;
    float sA = sc1[gch],     tA = sh1[gch];
    float sB = sc1[gch + 1], tB = sh1[gch + 1];
    v2f a; a.x = Wg[c]; a.y = Wg[c + 1];
    const float* bp = hbI + (size_t)c * SP;
#pragma unroll
    for (int t = 0; t < 4; ++t) {
      float r0 = fmaf(bp[t * 16],      sA, tA);
      float r1 = fmaf(bp[t * 16 + SP], sB, tB);
      r0 = r0 > 0.0f ? r0 : (__expf(r0) - 1.0f);   // ELU(alpha=1)
      r1 = r1 > 0.0f ? r1 : (__expf(r1) - 1.0f);
      v2f bb; bb.x = r0; bb.y = r1;
      acc[t] = wmma_f32(a, bb, acc[t]);
    }
  }

  float* ob = h2 + (size_t)(b * OUTC + g * COUTG + m0 + hl * 8) * SP + s0 + ln;
#pragma unroll
  for (int v = 0; v < 8; ++v) {
    float s = 0.0f, q = 0.0f;
#pragma unroll
    for (int t = 0; t < 4; ++t) {
      float val = acc[t][v];
      ob[(size_t)v * SP + t * 16] = val;
      s += val; q += val * val;
    }
#pragma unroll
    for (int off = 8; off; off >>= 1) {
      s += __shfl_xor(s, off, 16);
      q += __shfl_xor(q, off, 16);
    }
    if (ln == 0) {
      int ch = g * COUTG + m0 + hl * 8 + v;
      atomicAdd(&sum2[ch], s);
      atomicAdd(&sq2[ch], q);
    }
  }
}

// ---------------- adjacency: y[b,g,c,j,:] = BN2(h2)[b,g,c,j,:] @ L[g]  (K=62)
// per (b,g): rows = 96*256 = 24576, grid = B*G*192 blocks, 8 m-tiles/block
__global__ __launch_bounds__(256)
void adjacency_kernel(const float* __restrict__ h2, const float* __restrict__ L,
                      const float* __restrict__ sc2, const float* __restrict__ sh2,
                      float* __restrict__ y) {
  __shared__ float Lg[Kc * Kc];                 // 15,376 B in LDS
  const int bg = blockIdx.x / 192;
  const int mblk = blockIdx.x % 192;
  const int b = bg / Gc, g = bg % Gc;

  for (int i = threadIdx.x; i < Kc * Kc; i += 256)
    Lg[i] = L[g * Kc * Kc + i];
  __syncthreads();

  const int lane = threadIdx.x & 31;
  const int hl = lane >> 4, ln = lane & 15;
  const int r0 = (mblk * 8 + (threadIdx.x >> 5)) * 16;   // row tile base
  const size_t base = (size_t)(b * OUTC + g * COUTG) * SP;

  const int rowA = r0 + ln;
  const int cA = rowA >> 8;                    // channel within group (/256)
  const float sA = sc2[g * COUTG + cA];
  const float tA = sh2[g * COUTG + cA];
  const float* Arow = h2 + base + (size_t)rowA * Kc;

  v8f acc[4] = {};
#pragma unroll
  for (int kk = 0; kk < 16; ++kk) {            // K=62 padded to 64
    const int k = kk * 4 + hl * 2;
    v2f a;
    a.x = (k     < Kc) ? fmaf(Arow[k],     sA, tA) : 0.0f;
    a.y = (k + 1 < Kc) ? fmaf(Arow[k + 1], sA, tA) : 0.0f;
#pragma unroll
    for (int t = 0; t < 4; ++t) {
      const int p = t * 16 + ln;
      v2f bb;
      bb.x = (k     < Kc && p < Kc) ? Lg[k * Kc + p]       : 0.0f;
      bb.y = (k + 1 < Kc && p < Kc) ? Lg[(k + 1) * Kc + p] : 0.0f;
      acc[t] = wmma_f32(a, bb, acc[t]);
    }
  }

#pragma unroll
  for (int t = 0; t < 4; ++t) {
    const int p = t * 16 + ln;
    if (p < Kc) {
#pragma unroll
      for (int v = 0; v < 8; ++v) {
        const int rr = r0 + v + hl * 8;
        y[base + (size_t)rr * Kc + p] = acc[t][v];
      }
    }
  }
}

// ---------------------------------------------------------------- launcher
extern "C" void kernel_launch(void* const* d_in, const int* in_sizes, int n_in,
                              void* d_out, int out_size, void* d_ws, size_t ws_size,
                              hipStream_t stream) {
  (void)in_sizes; (void)n_in; (void)out_size; (void)ws_size;
  const float* x      = (const float*)d_in[0];
  const float* L      = (const float*)d_in[1];
  const float* W1     = (const float*)d_in[2];
  const float* W2     = (const float*)d_in[3];
  const float* gamma1 = (const float*)d_in[4];
  const float* beta1  = (const float*)d_in[5];
  const float* gamma2 = (const float*)d_in[6];
  const float* beta2  = (const float*)d_in[7];
  float* y  = (float*)d_out;
  float* ws = (float*)d_ws;

  float* h     = ws;
  float* h2    = h + HSZ;
  float* stats = h2 + HSZ;
  float* sum1 = stats;          float* sq1 = stats + 480;
  float* sum2 = stats + 960;    float* sq2 = stats + 1440;
  float* sc1  = stats + 1920;   float* sh1 = stats + 2400;
  float* sc2  = stats + 2880;   float* sh2 = stats + 3360;

  const float invN = 1.0f / (float)(Bc * SP);  // BN over B*J*K

  zero_stats_kernel<<<(1920 + 255) / 256, 256, 0, stream>>>(stats, 1920);

  conv1_kernel<<<7440, 256, 0, stream>>>(x, W1, h, sum1, sq1);
  bn_finalize_kernel<<<2, 256, 0, stream>>>(sum1, sq1, gamma1, beta1, sc1, sh1, invN);

  conv2_kernel<<<7440, 256, 0, stream>>>(h, W2, sc1, sh1, h2, sum2, sq2);
  bn_finalize_kernel<<<2, 256, 0, stream>>>(sum2, sq2, gamma2, beta2, sc2, sh2, invN);

  adjacency_kernel<<<Bc * Gc * 192, 256, 0, stream>>>(h2, L, sc2, sh2, y);
}